// Origin_HyperKA_9715216023648
// MI455X (gfx1250) — compile-verified
//
#include <hip/hip_runtime.h>
#include <math.h>

#define DIM 128
#define PROJ_EPS 1e-5f
#define MIN_NORM 1e-15f

typedef __attribute__((ext_vector_type(2))) float v2f;
typedef __attribute__((ext_vector_type(8))) float v8f;

// ---------------- wave-level helpers (wave32) ----------------

__device__ __forceinline__ float wave_sum(float v) {
#pragma unroll
  for (int off = 16; off >= 1; off >>= 1)
    v += __shfl_xor(v, off, 32);
  return v;
}

// dot over a row distributed as 4 floats per lane across 32 lanes
__device__ __forceinline__ float row_dot(const float a[4], const float b[4]) {
  float s = a[0] * b[0] + a[1] * b[1] + a[2] * b[2] + a[3] * b[3];
  return wave_sum(s);
}

__device__ __forceinline__ void hyp_proj(float r[4]) {
  float n = sqrtf(fmaxf(row_dot(r, r), 0.0f));
  const float maxn = 1.0f - PROJ_EPS;
  float scale = (n > maxn) ? (maxn / fmaxf(n, MIN_NORM)) : 1.0f;
#pragma unroll
  for (int i = 0; i < 4; ++i) r[i] *= scale;
}

__device__ __forceinline__ void exp_map_zero(float r[4]) {
  float n = fmaxf(sqrtf(fmaxf(row_dot(r, r), 0.0f)), MIN_NORM);
  float s = tanhf(n) / n;
#pragma unroll
  for (int i = 0; i < 4; ++i) r[i] *= s;
}

__device__ __forceinline__ void log_map_zero(float r[4]) {
  float n = fmaxf(sqrtf(fmaxf(row_dot(r, r), 0.0f)), MIN_NORM);
  float nc = fminf(fmaxf(n, MIN_NORM), 1.0f - PROJ_EPS);
  float s = atanhf(nc) / n;
#pragma unroll
  for (int i = 0; i < 4; ++i) r[i] *= s;
}

__device__ __forceinline__ void mobius_add(const float u[4], const float v[4],
                                           float out[4]) {
  float uu = row_dot(u, u);
  float vv = row_dot(v, v);
  float uv = row_dot(u, v);
  float cu = 1.0f + 2.0f * uv + vv;
  float cv = 1.0f - uu;
  float den = fmaxf(1.0f + 2.0f * uv + uu * vv, MIN_NORM);
  float inv = 1.0f / den;
#pragma unroll
  for (int i = 0; i < 4; ++i) out[i] = (cu * u[i] + cv * v[i]) * inv;
}

// ---------------- kernels ----------------

// x = hyperbolic_projection(ent) ; one wave per row
__global__ void __launch_bounds__(256) k_init_proj(const float* __restrict__ ent,
                                                   float* __restrict__ x, int N) {
  int wave = threadIdx.x >> 5, lane = threadIdx.x & 31;
  int row = blockIdx.x * (blockDim.x >> 5) + wave;
  if (row >= N) return;
  const float* src = ent + (size_t)row * DIM + lane * 4;
  float r[4];
#pragma unroll
  for (int i = 0; i < 4; ++i) r[i] = src[i];
  hyp_proj(r);
  float* dst = x + (size_t)row * DIM + lane * 4;
#pragma unroll
  for (int i = 0; i < 4; ++i) dst[i] = r[i];
}

// t = log_map_zero(x) ; one wave per row
__global__ void __launch_bounds__(256) k_logmap(const float* __restrict__ x,
                                                float* __restrict__ t, int N) {
  int wave = threadIdx.x >> 5, lane = threadIdx.x & 31;
  int row = blockIdx.x * (blockDim.x >> 5) + wave;
  if (row >= N) return;
  const float* src = x + (size_t)row * DIM + lane * 4;
  float r[4];
#pragma unroll
  for (int i = 0; i < 4; ++i) r[i] = src[i];
  log_map_zero(r);
  float* dst = t + (size_t)row * DIM + lane * 4;
#pragma unroll
  for (int i = 0; i < 4; ++i) dst[i] = r[i];
}

// M = T @ W  : [N,128] x [128,128] fp32 via V_WMMA_F32_16X16X4_F32
// Block: 256 threads = 8 waves. Each wave owns one 16x16 output tile:
// rows [blockIdx.x*16, +16), cols [wave*16, +16). K loop of 32 WMMAs.
__global__ void __launch_bounds__(256) k_gemm_wmma(const float* __restrict__ T,
                                                   const float* __restrict__ W,
                                                   float* __restrict__ M, int N) {
  int wave = threadIdx.x >> 5;
  int lane = threadIdx.x & 31;
  int rowBase = blockIdx.x * 16;
  int colBase = wave * 16;

  int r = lane & 15;          // A-row / B-col within tile
  int khalf = (lane >> 4) * 2;  // lanes 0-15 -> K{0,1}; lanes 16-31 -> K{2,3}

  int arow = rowBase + r;
  if (arow >= N) arow = N - 1;  // clamp (N multiple of 16 in practice)
  const float* trow = T + (size_t)arow * DIM;

  v8f c = {};
#pragma unroll
  for (int k = 0; k < DIM; k += 4) {
    v2f a, b;
    a.x = trow[k + khalf];
    a.y = trow[k + khalf + 1];
    b.x = W[(size_t)(k + khalf) * DIM + colBase + r];
    b.y = W[(size_t)(k + khalf + 1) * DIM + colBase + r];
    // D = A(16x4) * B(4x16) + C ; 8 args: neg_a, A, neg_b, B, c_mod, C, reuse_a, reuse_b
    c = __builtin_amdgcn_wmma_f32_16x16x4_f32(false, a, false, b, (short)0, c,
                                              false, false);
  }

  // C/D layout: VGPR v, lanes 0-15 -> (M=v, N=lane); lanes 16-31 -> (M=v+8, N=lane-16)
  int n = lane & 15;
  int mOff = (lane >> 4) * 8;
#pragma unroll
  for (int v = 0; v < 8; ++v) {
    int orow = rowBase + mOff + v;
    if (orow < N) M[(size_t)orow * DIM + colBase + n] = c[v];
  }
}

__global__ void __launch_bounds__(256) k_zero(float* __restrict__ p, long long n) {
  long long i = blockIdx.x * (long long)blockDim.x + threadIdx.x;
  if (i < n) p[i] = 0.0f;
}

// agg[rows[e]] += M[cols[e]] * vals[e] ; one 32-lane group per edge, float4 gathers
__global__ void __launch_bounds__(256) k_spmm(const float* __restrict__ M,
                                              const int* __restrict__ rows,
                                              const int* __restrict__ cols,
                                              const float* __restrict__ vals,
                                              float* __restrict__ agg, int E) {
  long long gid = blockIdx.x * (long long)blockDim.x + threadIdx.x;
  int e = (int)(gid >> 5);
  int lane = (int)(gid & 31);
  if (e >= E) return;
  int drow = rows[e];
  int srow = cols[e];
  float v = vals[e];
  float4 m4 = ((const float4*)(M + (size_t)srow * DIM))[lane];
  float* dst = agg + (size_t)drow * DIM + lane * 4;
  atomicAdd(dst + 0, m4.x * v);
  atomicAdd(dst + 1, m4.y * v);
  atomicAdd(dst + 2, m4.z * v);
  atomicAdd(dst + 3, m4.w * v);
}

// h = proj(exp(agg)); bh = proj(exp(bias)); h = proj(h (+) bh);
// if act: h = proj(exp(tanh(log(h)))); x_out = proj(h (+) x_in)
__global__ void __launch_bounds__(256) k_epilogue(const float* __restrict__ agg,
                                                  const float* __restrict__ bias_l,
                                                  const float* __restrict__ x_in,
                                                  float* __restrict__ x_out,
                                                  int N, int use_act) {
  int wave = threadIdx.x >> 5, lane = threadIdx.x & 31;
  int row = blockIdx.x * (blockDim.x >> 5) + wave;
  if (row >= N) return;

  float h[4], xr[4], bh[4], tmp[4];
  const float* asrc = agg + (size_t)row * DIM + lane * 4;
  const float* xsrc = x_in + (size_t)row * DIM + lane * 4;
  const float* bsrc = bias_l + lane * 4;
#pragma unroll
  for (int i = 0; i < 4; ++i) {
    h[i] = asrc[i];
    xr[i] = xsrc[i];
    bh[i] = bsrc[i];
  }

  exp_map_zero(h);
  hyp_proj(h);

  exp_map_zero(bh);
  hyp_proj(bh);

  mobius_add(h, bh, tmp);
  hyp_proj(tmp);
#pragma unroll
  for (int i = 0; i < 4; ++i) h[i] = tmp[i];

  if (use_act) {
    log_map_zero(h);
#pragma unroll
    for (int i = 0; i < 4; ++i) h[i] = tanhf(h[i]);
    exp_map_zero(h);
    hyp_proj(h);
  }

  mobius_add(h, xr, tmp);
  hyp_proj(tmp);

  float* dst = x_out + (size_t)row * DIM + lane * 4;
#pragma unroll
  for (int i = 0; i < 4; ++i) dst[i] = tmp[i];
}

// ---------------- host orchestration ----------------

extern "C" void kernel_launch(void* const* d_in, const int* in_sizes, int n_in,
                              void* d_out, int out_size, void* d_ws, size_t ws_size,
                              hipStream_t stream) {
  const float* ent  = (const float*)d_in[0];
  const float* W    = (const float*)d_in[1];
  const float* bias = (const float*)d_in[2];
  const int* rows   = (const int*)d_in[3];
  const int* cols   = (const int*)d_in[4];
  const float* vals = (const float*)d_in[5];

  int N = in_sizes[0] / DIM;
  int L = in_sizes[2] / DIM;
  int E = in_sizes[3];
  float* out = (float*)d_out;

  size_t rowElems = (size_t)N * DIM;
  float* bufT = (float*)d_ws;        // t, then reused as agg
  float* bufM = bufT + rowElems;     // m = t @ W
  float* bufX = bufM + rowElems;     // current x

  dim3 blk(256);
  int rowsPerBlock = 256 / 32;  // 8 waves per block, one row each
  int gridRows = (N + rowsPerBlock - 1) / rowsPerBlock;
  int gridGemm = (N + 15) / 16;
  long long nd = (long long)N * DIM;
  int gridZero = (int)((nd + 255) / 256);
  long long spmmThreads = (long long)E * 32;
  int gridSpmm = (int)((spmmThreads + 255) / 256);

  // x = proj(ent)
  k_init_proj<<<gridRows, blk, 0, stream>>>(ent, bufX, N);

  for (int l = 0; l < L; ++l) {
    const float* Wl = W + (size_t)l * DIM * DIM;
    const float* bl = bias + (size_t)l * DIM;
    int use_act = (l < L - 1) ? 1 : 0;
    float* dst = (l == L - 1) ? out : bufX;

    // t = log_map_zero(x)
    k_logmap<<<gridRows, blk, 0, stream>>>(bufX, bufT, N);
    // m = t @ W_l   (WMMA fp32)
    k_gemm_wmma<<<gridGemm, blk, 0, stream>>>(bufT, Wl, bufM, N);
    // agg = 0 (reuse bufT)
    k_zero<<<gridZero, blk, 0, stream>>>(bufT, nd);
    // agg += adj @ m
    k_spmm<<<gridSpmm, blk, 0, stream>>>(bufM, rows, cols, vals, bufT, E);
    // hyperbolic epilogue + residual; in-place per-row update of x is safe
    k_epilogue<<<gridRows, blk, 0, stream>>>(bufT, bl, bufX, dst, N, use_act);
  }
}